// ChannelNormalization_80616536146731
// MI455X (gfx1250) — compile-verified
//
#include <hip/hip_runtime.h>

// ---------------------------------------------------------------------------
// ChannelNormalization (instance-norm over H*W per channel, unbiased variance)
// x: (1, C, H, W) fp32; out = (x - mean_c) / sqrt(var_c + 1e-5) + beta[c]
//
// MI455X reasoning: 256MB in / 256MB out, ~0.5 GFLOP -> pure bandwidth problem
// (~33us floor at 23.3 TB/s for read+read+write). One 1024-thread block per
// channel, fused two-pass so the 1MB channel re-read hits L2. Cross-lane
// reductions use the exact-fp32 V_WMMA_F32_16X16X4_F32 path (CDNA5 WMMA) so
// the matrix pipe does the (sum, sumsq) tree reduce with zero precision loss.
// Streaming loops are unrolled (8x / 4x) so each wave keeps multiple
// global_load_b128 requests in flight; pass-2 traffic is non-temporal so a
// consumed channel stops occupying L2.
// ---------------------------------------------------------------------------

typedef __attribute__((ext_vector_type(2))) float v2f;
typedef __attribute__((ext_vector_type(4))) float v4f;
typedef __attribute__((ext_vector_type(8))) float v8f;

// Reduce (s, q) across the 32 lanes of a wave using V_WMMA_F32_16X16X4_F32.
// A-matrix layout (32-bit, 16x4): lane L, VGPR j holds A[L%16][2*(L/16)+j],
// so a = (s, q) gives A[m][0]=s_m, A[m][1]=q_m, A[m][2]=s_{m+16}, A[m][3]=q_{m+16}.
// B-matrix layout (4x16): lane L, VGPR j holds B[2*(L/16)+j][L%16].
// Selector B: column 0 = [1,0,1,0]^T (picks sums), column 1 = [0,1,0,1]^T
// (picks sumsqs), all other columns zero. Then
//   D[m][0] = s_m + s_{m+16},   D[m][1] = q_m + q_{m+16}
// Each lane sums its 8 accumulator VGPRs (rows m = 8*(L/16)..8*(L/16)+7 of
// column L%16); lane0/lane16 then hold the two halves of the total sum and
// lane1/lane17 the two halves of the total sumsq. One xor-16 shuffle finishes.
__device__ __forceinline__ float2 wave_reduce_pair(float s, float q, int lane) {
  v2f a;
  a.x = s;
  a.y = q;
  const int n = lane & 15;
  v2f b;
  b.x = (n == 0) ? 1.0f : 0.0f;
  b.y = (n == 1) ? 1.0f : 0.0f;
  v8f c = {};
  // (neg_a, A, neg_b, B, c_mod, C, reuse_a, reuse_b)
  c = __builtin_amdgcn_wmma_f32_16x16x4_f32(false, a, false, b, (short)0, c,
                                            false, false);
  float t = ((c[0] + c[1]) + (c[2] + c[3])) + ((c[4] + c[5]) + (c[6] + c[7]));
  t += __shfl_xor(t, 16, 32);          // combine lane halves: lane0=S, lane1=Q
  const float S = __shfl(t, 0, 32);
  const float Q = __shfl(t, 1, 32);
  return make_float2(S, Q);
}

__global__ __launch_bounds__(1024) void channelnorm_kernel(
    const float* __restrict__ x, const float* __restrict__ beta,
    float* __restrict__ out, int HW) {
  const int c    = blockIdx.x;
  const int tid  = (int)threadIdx.x;
  const int lane = tid & 31;
  const int wid  = tid >> 5;
  const int nthr = (int)blockDim.x;        // 1024 -> 32 waves
  const int n4   = HW >> 2;                // float4 elements per channel

  const v4f* __restrict__ xrow = (const v4f*)(x + (size_t)c * (size_t)HW);
  v4f* __restrict__ orow       = (v4f*)(out + (size_t)c * (size_t)HW);

  // ---- Pass 1: streaming fp32 sum / sum-of-squares. Unroll 8x so each wave
  // has up to 8 global_load_b128 outstanding (LOADcnt-batched by the
  // scheduler); regular-temporal so the channel stays in L2 for pass 2. ----
  float s = 0.0f, q = 0.0f;
#pragma unroll 8
  for (int i = tid; i < n4; i += nthr) {
    const v4f v = xrow[i];
    s += (v.x + v.y) + (v.z + v.w);
    q = fmaf(v.x, v.x, q);
    q = fmaf(v.y, v.y, q);
    q = fmaf(v.z, v.z, q);
    q = fmaf(v.w, v.w, q);
  }

  __shared__ float2 wave_part[32];
  __shared__ float  stat[2];

  // Wave-level reduce via fp32 WMMA (EXEC is full here: uniform control flow).
  const float2 wr = wave_reduce_pair(s, q, lane);
  if (lane == 0) wave_part[wid] = wr;
  __syncthreads();

  // Block-level reduce: wave 0 reduces the 32 per-wave partials, same WMMA path.
  if (wid == 0) {
    const int nw = nthr >> 5;
    const float2 p = (lane < nw) ? wave_part[lane] : make_float2(0.0f, 0.0f);
    const float2 tot = wave_reduce_pair(p.x, p.y, lane);
    if (lane == 0) {
      const float N    = (float)HW;
      const float mean = tot.x / N;
      const float var  = (tot.y - tot.x * tot.x / N) / (N - 1.0f);  // unbiased
      stat[0] = mean;
      stat[1] = 1.0f / sqrtf(var + 1e-5f);
    }
  }
  __syncthreads();

  const float mu = stat[0];
  const float rs = stat[1];
  const float bt = beta[c];

  // ---- Pass 2: normalize. Data is dead after this: non-temporal load/store
  // so this channel stops occupying L2 once consumed. Unroll 4x for MLP. ----
#pragma unroll 4
  for (int i = tid; i < n4; i += nthr) {
    const v4f v = __builtin_nontemporal_load(xrow + i);
    v4f o;
    o.x = fmaf(v.x - mu, rs, bt);
    o.y = fmaf(v.y - mu, rs, bt);
    o.z = fmaf(v.z - mu, rs, bt);
    o.w = fmaf(v.w - mu, rs, bt);
    __builtin_nontemporal_store(o, orow + i);
  }
}

extern "C" void kernel_launch(void* const* d_in, const int* in_sizes, int n_in,
                              void* d_out, int out_size, void* d_ws,
                              size_t ws_size, hipStream_t stream) {
  const float* x    = (const float*)d_in[0];   // (1, C, H, W) fp32
  // d_in[1] = gamma: unused by the reference in this mode
  const float* beta = (const float*)d_in[2];   // (C,) fp32
  float* out        = (float*)d_out;

  const int C  = in_sizes[2];                  // 256
  const int HW = in_sizes[0] / C;              // 512*512 = 262144 (mult. of 4)

  channelnorm_kernel<<<dim3(C), dim3(1024), 0, stream>>>(x, beta, out, HW);
}